// Model_39676907881751
// MI455X (gfx1250) — compile-verified
//
#include <hip/hip_runtime.h>

typedef __attribute__((ext_vector_type(2))) float v2f;
typedef __attribute__((ext_vector_type(8))) float v8f;

constexpr int kB    = 4;
constexpr int kS    = 4096;
constexpr int kDin  = 3;
constexpr int kDk   = 5;
constexpr int kPad  = 8;   // padded row stride for q/k/v workspace

#define NEG_INF (-__builtin_huge_valf())
// (1/sqrt(5)) * log2(e): logits come out of WMMA already in log2-units.
#define QSCALE 0.6451496129472319f

// ---------------------------------------------------------------------------
// DPP16 cross-lane helpers (pure VALU). On wave32 a DPP row is 16 lanes ==
// our reduction group. xor1, xor2, then half/full mirror (== xor4/xor8 once
// sub-groups are uniform) -> full 16-lane reduction broadcast to all lanes.
// ---------------------------------------------------------------------------
#define DPP_QUAD_XOR1  0xB1   // quad_perm(1,0,3,2)
#define DPP_QUAD_XOR2  0x4E   // quad_perm(2,3,0,1)
#define DPP_HALF_MIRR  0x141  // row_half_mirror
#define DPP_ROW_MIRR   0x140  // row_mirror

template <int CTRL>
__device__ __forceinline__ float dpp_mov(float x) {
  return __builtin_bit_cast(
      float, __builtin_amdgcn_update_dpp(0, __builtin_bit_cast(int, x), CTRL,
                                         0xF, 0xF, true));
}

__device__ __forceinline__ float reduce16_max(float x) {
  x = fmaxf(x, dpp_mov<DPP_QUAD_XOR1>(x));
  x = fmaxf(x, dpp_mov<DPP_QUAD_XOR2>(x));
  x = fmaxf(x, dpp_mov<DPP_HALF_MIRR>(x));
  x = fmaxf(x, dpp_mov<DPP_ROW_MIRR>(x));
  return x;
}

__device__ __forceinline__ float reduce16_sum(float x) {
  x += dpp_mov<DPP_QUAD_XOR1>(x);
  x += dpp_mov<DPP_QUAD_XOR2>(x);
  x += dpp_mov<DPP_HALF_MIRR>(x);
  x += dpp_mov<DPP_ROW_MIRR>(x);
  return x;
}

// ---------------------------------------------------------------------------
// q/k/v projections -> padded [B*S, 8] rows (cols 5..7 = 0) so the attention
// kernel can use aligned b64 fragment loads with zero conditionals.
// ---------------------------------------------------------------------------
__global__ __launch_bounds__(256) void proj_kernel(
    const float* __restrict__ x,
    const float* __restrict__ Wq, const float* __restrict__ bq,
    const float* __restrict__ Wk, const float* __restrict__ bk,
    const float* __restrict__ Wv, const float* __restrict__ bv,
    float* __restrict__ qo, float* __restrict__ ko, float* __restrict__ vo) {
  int i = blockIdx.x * blockDim.x + threadIdx.x;
  if (i >= kB * kS) return;
  const float x0 = x[i * kDin + 0];
  const float x1 = x[i * kDin + 1];
  const float x2 = x[i * kDin + 2];

  float qv[kDk], kv[kDk], vv[kDk];
#pragma unroll
  for (int o = 0; o < kDk; ++o) {
    qv[o] = fmaf(Wq[o * kDin + 0], x0,
            fmaf(Wq[o * kDin + 1], x1, fmaf(Wq[o * kDin + 2], x2, bq[o])));
    kv[o] = fmaf(Wk[o * kDin + 0], x0,
            fmaf(Wk[o * kDin + 1], x1, fmaf(Wk[o * kDin + 2], x2, bk[o])));
    vv[o] = fmaf(Wv[o * kDin + 0], x0,
            fmaf(Wv[o * kDin + 1], x1, fmaf(Wv[o * kDin + 2], x2, bv[o])));
  }
  // two aligned 16B stores per row (row start is 32B aligned)
  *(float4*)(qo + (size_t)i * kPad)     = make_float4(qv[0], qv[1], qv[2], qv[3]);
  *(float4*)(qo + (size_t)i * kPad + 4) = make_float4(qv[4], 0.f, 0.f, 0.f);
  *(float4*)(ko + (size_t)i * kPad)     = make_float4(kv[0], kv[1], kv[2], kv[3]);
  *(float4*)(ko + (size_t)i * kPad + 4) = make_float4(kv[4], 0.f, 0.f, 0.f);
  *(float4*)(vo + (size_t)i * kPad)     = make_float4(vv[0], vv[1], vv[2], vv[3]);
  *(float4*)(vo + (size_t)i * kPad + 4) = make_float4(vv[4], 0.f, 0.f, 0.f);
}

// ---------------------------------------------------------------------------
// Fused attention, two-pass softmax. One wave32 per 16-query tile.
// Logits tile = chained f32 WMMA (K=5 padded to 8 = 2x4), pre-scaled so
// exp == exp2. Pass 1: full-row max. Pass 2 split at the causal diagonal.
// Padded-row b64 loads give both wave halves their K-pair with no branches.
// ---------------------------------------------------------------------------
__global__ __launch_bounds__(32) void attn_kernel(
    const float* __restrict__ q, const float* __restrict__ k,
    const float* __restrict__ v, float* __restrict__ out) {
  const int ntile  = kS / 16;
  const int b      = blockIdx.x / ntile;
  const int q_base = (blockIdx.x % ntile) * 16;
  const int lane   = threadIdx.x & 31;
  const int r      = lane & 15;
  const int hi     = lane >> 4;

  const float* __restrict__ qb = q + (size_t)b * kS * kPad;
  const float* __restrict__ kb = k + (size_t)b * kS * kPad;
  const float* __restrict__ vb = v + (size_t)b * kS * kPad;

  // A fragments: aligned b64 loads from padded row (zeros make pads exact).
  const float* qrow = qb + (size_t)(q_base + r) * kPad + 2 * hi;
  v2f a0 = *(const v2f*)(qrow)     * QSCALE;  // K{0,1} / K{2,3}
  v2f a1 = *(const v2f*)(qrow + 4) * QSCALE;  // K{4,5} / K{6,7} (pads = 0)

  const float* kcol = kb + (size_t)r * kPad + 2 * hi;  // + s_base*kPad per tile

  // ---------------- Pass 1: full-row max (log2-units) ----------------
  float Mx[8];
#pragma unroll
  for (int j = 0; j < 8; ++j) Mx[j] = NEG_INF;

  for (int s_base = 0; s_base < kS; s_base += 16) {
    const float* kr = kcol + (size_t)s_base * kPad;
    v2f b0 = *(const v2f*)(kr);
    v2f b1 = *(const v2f*)(kr + 4);

    v8f c = {};
    c = __builtin_amdgcn_wmma_f32_16x16x4_f32(false, a0, false, b0,
                                              (short)0, c, false, false);
    c = __builtin_amdgcn_wmma_f32_16x16x4_f32(false, a1, false, b1,
                                              (short)0, c, false, false);
#pragma unroll
    for (int j = 0; j < 8; ++j) Mx[j] = fmaxf(Mx[j], c[j]);
  }
#pragma unroll
  for (int j = 0; j < 8; ++j) Mx[j] = reduce16_max(Mx[j]);  // row-uniform

  // ---------------- Pass 2 ----------------
  float Ls[8], acc[8][kDk];
#pragma unroll
  for (int j = 0; j < 8; ++j) {
    Ls[j] = 0.0f;
#pragma unroll
    for (int d = 0; d < kDk; ++d) acc[j][d] = 0.0f;
  }

  // (a) fully unmasked tiles (all cols < all rows of this tile)
  for (int s_base = 0; s_base < q_base; s_base += 16) {
    const float* kr = kcol + (size_t)s_base * kPad;
    v2f b0 = *(const v2f*)(kr);
    v2f b1 = *(const v2f*)(kr + 4);
    const float* vr = vb + (size_t)(s_base + r) * kPad;
    float vv[kDk];
#pragma unroll
    for (int d = 0; d < kDk; ++d) vv[d] = vr[d];

    v8f c = {};
    c = __builtin_amdgcn_wmma_f32_16x16x4_f32(false, a0, false, b0,
                                              (short)0, c, false, false);
    c = __builtin_amdgcn_wmma_f32_16x16x4_f32(false, a1, false, b1,
                                              (short)0, c, false, false);
#pragma unroll
    for (int j = 0; j < 8; ++j) {
      const float p = __builtin_amdgcn_exp2f(c[j] - Mx[j]);
      Ls[j] += p;
#pragma unroll
      for (int d = 0; d < kDk; ++d) acc[j][d] = fmaf(p, vv[d], acc[j][d]);
    }
  }

  // (b) diagonal tile: mask gates only the numerator
  {
    const float* kr = kcol + (size_t)q_base * kPad;
    v2f b0 = *(const v2f*)(kr);
    v2f b1 = *(const v2f*)(kr + 4);
    const float* vr = vb + (size_t)(q_base + r) * kPad;
    float vv[kDk];
#pragma unroll
    for (int d = 0; d < kDk; ++d) vv[d] = vr[d];

    v8f c = {};
    c = __builtin_amdgcn_wmma_f32_16x16x4_f32(false, a0, false, b0,
                                              (short)0, c, false, false);
    c = __builtin_amdgcn_wmma_f32_16x16x4_f32(false, a1, false, b1,
                                              (short)0, c, false, false);
#pragma unroll
    for (int j = 0; j < 8; ++j) {
      const float p = __builtin_amdgcn_exp2f(c[j] - Mx[j]);
      Ls[j] += p;                                     // denominator: unmasked
      const float pm = (r <= j + 8 * hi) ? p : 0.0f;  // col <= row in tile
#pragma unroll
      for (int d = 0; d < kDk; ++d) acc[j][d] = fmaf(pm, vv[d], acc[j][d]);
    }
  }

  // (c) fully masked tiles: denominator only (mask applied AFTER softmax)
  for (int s_base = q_base + 16; s_base < kS; s_base += 16) {
    const float* kr = kcol + (size_t)s_base * kPad;
    v2f b0 = *(const v2f*)(kr);
    v2f b1 = *(const v2f*)(kr + 4);

    v8f c = {};
    c = __builtin_amdgcn_wmma_f32_16x16x4_f32(false, a0, false, b0,
                                              (short)0, c, false, false);
    c = __builtin_amdgcn_wmma_f32_16x16x4_f32(false, a1, false, b1,
                                              (short)0, c, false, false);
#pragma unroll
    for (int j = 0; j < 8; ++j)
      Ls[j] += __builtin_amdgcn_exp2f(c[j] - Mx[j]);
  }

  // ---------------- Merge lane partials + store ----------------
#pragma unroll
  for (int j = 0; j < 8; ++j) {
    const float lsum = reduce16_sum(Ls[j]);
    const float inv  = 1.0f / lsum;
    const int   row  = q_base + j + 8 * hi;
    float red[kDk];
#pragma unroll
    for (int d = 0; d < kDk; ++d)
      red[d] = reduce16_sum(acc[j][d]) * inv;
    if (r == 0) {
#pragma unroll
      for (int d = 0; d < kDk; ++d)
        out[((size_t)b * kS + row) * kDk + d] = red[d];
    }
  }
}

// ---------------------------------------------------------------------------
extern "C" void kernel_launch(void* const* d_in, const int* in_sizes, int n_in,
                              void* d_out, int out_size, void* d_ws,
                              size_t ws_size, hipStream_t stream) {
  const float* x  = (const float*)d_in[0];
  const float* Wq = (const float*)d_in[1];
  const float* bq = (const float*)d_in[2];
  const float* Wk = (const float*)d_in[3];
  const float* bk = (const float*)d_in[4];
  const float* Wv = (const float*)d_in[5];
  const float* bv = (const float*)d_in[6];
  float* out = (float*)d_out;

  float* qws = (float*)d_ws;                     // [B*S*8] padded
  float* kws = qws + (size_t)kB * kS * kPad;     // [B*S*8] padded
  float* vws = kws + (size_t)kB * kS * kPad;     // [B*S*8] padded (~1.5 MB)

  const int total = kB * kS;
  proj_kernel<<<(total + 255) / 256, 256, 0, stream>>>(x, Wq, bq, Wk, bk, Wv,
                                                       bv, qws, kws, vws);
  attn_kernel<<<kB * (kS / 16), 32, 0, stream>>>(qws, kws, vws, out);
}